// Dense_RBS_density_3D_31963146617516
// MI455X (gfx1250) — compile-verified
//
#include <hip/hip_runtime.h>
#include <hip/hip_bf16.h>
#include <math.h>

// ---------------------------------------------------------------------------
// RBS density evolution: out[b] = M rho[b] M^T, M = U38 ... U0 (39 Givens-block
// rotations over the Hamming-weight-2 basis of 40 qubits, D = 780).
// Build M once (sparse rotations on identity), then two fp32 WMMA GEMM passes.
// D = 780 = 195*4 exactly, so the 16x16x4 K-loop needs no K padding; M/N edge
// tiles are handled by index clamping (preamble) + masked C store (epilogue);
// interior waves (82%) take a branch-free unmasked store path.
// ---------------------------------------------------------------------------

#define D      780
#define NGATES 39
#define NT     7     // N tiles per wave (7 strips of 7 tiles cover 49 tiles)
#define MTW    2     // M tiles per wave (B fragments shared across both)

typedef float v2f __attribute__((ext_vector_type(2)));
typedef float v8f __attribute__((ext_vector_type(8)));

// index of basis state {p,q}, p<q, in lexicographic combinations(40,2)
__device__ __forceinline__ int pair_index(int p, int q) {
    return p * 39 - (p * (p - 1)) / 2 + (q - p - 1);
}

__global__ void build_M_identity(float* __restrict__ M) {
    int idx = blockIdx.x * blockDim.x + threadIdx.x;
    if (idx < D * D) {
        int i = idx / D, j = idx % D;
        M[idx] = (i == j) ? 1.0f : 0.0f;
    }
}

// Sequentially apply the 39 gates to M (M <- U_g M). One block; each gate is
// 38 independent row-pair rotations x 780 columns, barrier per gate.
__global__ __launch_bounds__(1024) void apply_gates_to_M(
    float* __restrict__ M, const float* __restrict__ angles) {
    for (int g = 0; g < NGATES; ++g) {
        float th = angles[g];
        float c = cosf(th), s = sinf(th);
        int a = g, b = g + 1;
        for (int w = threadIdx.x; w < 38 * D; w += blockDim.x) {
            int p   = w / D;             // which partner qubit
            int col = w % D;
            int x = (p < a) ? p : p + 2; // partner in {0..39} \ {a, a+1}
            int i = (x < a) ? pair_index(x, a) : pair_index(a, x); // state has a
            int j = (x < b) ? pair_index(x, b) : pair_index(b, x); // state has b
            float mi = M[i * D + col];
            float mj = M[j * D + col];
            // U[i,i]=c, U[i,j]=-s, U[j,i]=s, U[j,j]=c
            M[i * D + col] = c * mi - s * mj;
            M[j * D + col] = s * mi + c * mj;
        }
        __syncthreads();
    }
}

__global__ void transpose_M(const float* __restrict__ M, float* __restrict__ MT) {
    int idx = blockIdx.x * blockDim.x + threadIdx.x;
    if (idx < D * D) {
        int i = idx / D, j = idx % D;
        MT[j * D + i] = M[idx];
    }
}

// Batched GEMM C[b] = A[b?] * B[b?] via V_WMMA_F32_16X16X4_F32.
// One wave per block; each wave produces a 32x112 patch (2 M-tiles x 7 N-tiles,
// 14 accumulators). Per k-step: 2 b64 A loads + 14 b32 B loads (shared across
// the 2 M-tiles) + 14 WMMAs. All indices clamped in the preamble.
__global__ __launch_bounds__(32) void rbs_gemm_wmma(
    const float* __restrict__ Abase, long aStride,
    const float* __restrict__ Bbase, long bStride,
    float*       __restrict__ Cbase, long cStride) {
    const int lane   = threadIdx.x;   // 0..31
    const int nStrip = blockIdx.x;    // 0..6
    const int mGroup = blockIdx.y;    // 0..24  (covers 50 M tiles; tile 49 fully masked)
    const int batch  = blockIdx.z;    // 0..63

    const float* A = Abase + (long)batch * aStride;
    const float* B = Bbase + (long)batch * bStride;
    float*       C = Cbase + (long)batch * cStride;

    // A fragment (16x4 f32, ISA 7.12.2): lanes 0-15 hold K={0,1}, lanes 16-31
    // hold K={2,3}; row = lane&15. Rows clamped (edge garbage is store-masked).
    const int lrow  = lane & 15;
    const int khalf = (lane >> 4) * 2;                 // 0 or 2
    const int arow0 = min(mGroup * 32 + lrow,      D - 1);
    const int arow1 = min(mGroup * 32 + 16 + lrow, D - 1);
    const int aOff0 = arow0 * D + khalf;               // per-lane, loop-invariant
    const int aOff1 = arow1 * D + khalf;

    // B fragment (4x16 f32): VGPR0 = row K=khalf, VGPR1 = row K=khalf+1; col = lane&15.
    const int n0 = nStrip * (16 * NT);
    int bOff[NT];
#pragma unroll
    for (int t = 0; t < NT; ++t)
        bOff[t] = khalf * D + min(n0 + t * 16 + lrow, D - 1);  // clamped cols

    v8f acc[MTW][NT] = {};

    for (int k0 = 0; k0 < D; k0 += 4) {                // 195 exact steps
        v2f a0 = *(const v2f*)&A[aOff0];
        v2f a1 = *(const v2f*)&A[aOff1];
        v2f bf[NT];
#pragma unroll
        for (int t = 0; t < NT; ++t) {
            bf[t].x = B[bOff[t]];
            bf[t].y = B[bOff[t] + D];
        }
#pragma unroll
        for (int t = 0; t < NT; ++t) {
            acc[0][t] = __builtin_amdgcn_wmma_f32_16x16x4_f32(
                false, a0, false, bf[t], (short)0, acc[0][t], false, false);
            acc[1][t] = __builtin_amdgcn_wmma_f32_16x16x4_f32(
                false, a1, false, bf[t], (short)0, acc[1][t], false, false);
        }
        A += 4;          // uniform scalar bump (K advances by 4)
        B += 4 * D;
    }

    // C/D layout (ISA 7.12.2): VGPR r, lanes 0-15 -> M=r, lanes 16-31 -> M=8+r
    const int rb   = mGroup * 32 + (lane >> 4) * 8;
    const int col0 = lane & 15;
    if (mGroup < 24 && nStrip < 6) {
        // Interior wave (82% of waves): whole 32x112 patch in-bounds, no masks.
#pragma unroll
        for (int m = 0; m < MTW; ++m) {
#pragma unroll
            for (int t = 0; t < NT; ++t) {
                const int n = n0 + t * 16 + col0;
#pragma unroll
                for (int r = 0; r < 8; ++r)
                    C[(long)(rb + m * 16 + r) * D + n] = acc[m][t][r];
            }
        }
    } else {
        // Edge wave: per-element masked stores.
#pragma unroll
        for (int m = 0; m < MTW; ++m) {
#pragma unroll
            for (int t = 0; t < NT; ++t) {
                const int n = n0 + t * 16 + col0;
#pragma unroll
                for (int r = 0; r < 8; ++r) {
                    const int rr = rb + m * 16 + r;
                    if (rr < D && n < D)
                        C[(long)rr * D + n] = acc[m][t][r];
                }
            }
        }
    }
}

extern "C" void kernel_launch(void* const* d_in, const int* in_sizes, int n_in,
                              void* d_out, int out_size, void* d_ws, size_t ws_size,
                              hipStream_t stream) {
    const float* rho    = (const float*)d_in[0];  // [64, 780, 780]
    const float* angles = (const float*)d_in[1];  // [39]
    // d_in[2..4] = A,B,C structural tensors; structure reproduced analytically.
    float* out = (float*)d_out;                   // [64, 780, 780]

    float* M   = (float*)d_ws;                    // [780, 780]
    float* MT  = M  + D * D;                      // [780, 780]
    float* tmp = MT + D * D;                      // [64, 780, 780]

    const int totalM = D * D;
    build_M_identity<<<(totalM + 255) / 256, 256, 0, stream>>>(M);
    apply_gates_to_M<<<1, 1024, 0, stream>>>(M, angles);
    transpose_M<<<(totalM + 255) / 256, 256, 0, stream>>>(M, MT);

    dim3 grid(NT, 25, 64);                        // 7 N-strips x 25 M-groups x 64 batches

    // pass 1: tmp[b] = M @ rho[b]   (A shared, B batched)
    rbs_gemm_wmma<<<grid, 32, 0, stream>>>(
        M,   0L,
        rho, (long)D * D,
        tmp, (long)D * D);

    // pass 2: out[b] = tmp[b] @ M^T (A batched, B = M^T shared)
    rbs_gemm_wmma<<<grid, 32, 0, stream>>>(
        tmp, (long)D * D,
        MT,  0L,
        out, (long)D * D);
}